// FaissKNNModule_61684320305634
// MI455X (gfx1250) — compile-verified
//
#include <hip/hip_runtime.h>

// ---- problem constants (match reference) ----
constexpr int KNN  = 50;      // neighbors
constexpr int CLS  = 1000;    // classes
constexpr int Q    = 2048;    // queries
constexpr int N    = 100000;  // gallery rows
constexpr int D    = 1024;    // feature dim

// ---- tiling ----
constexpr int QB      = 16;        // queries per workgroup (one WMMA M tile)
constexpr int NT      = 128;       // gallery rows per tile (8 waves x 16)
constexpr int DC      = 64;        // D chunk staged in LDS
constexpr int NCH     = D / DC;    // 16 chunks per tile
constexpr int THREADS = 256;       // 8 wave32
constexpr int QSTR    = D  + 4;    // query LDS row stride (bank spread)
constexpr int GSTR    = DC + 4;    // gallery LDS row stride
constexpr int SSTR    = NT + 4;    // score LDS row stride

typedef float v2f __attribute__((ext_vector_type(2)));
typedef float v8f __attribute__((ext_vector_type(8)));
typedef float f4  __attribute__((ext_vector_type(4)));

// pointee type expected by the async-to-LDS b128 builtin (from its diagnostic):
// "__attribute__((__vector_size__(4 * sizeof(int)))) int *"
typedef int vsi4 __attribute__((vector_size(16)));
using g128p = __attribute__((address_space(1))) vsi4*;   // global (AS1)
using l128p = __attribute__((address_space(3))) vsi4*;   // LDS (AS3)

#if __has_builtin(__builtin_amdgcn_global_load_async_to_lds_b128)
#define HAS_ASYNC_LDS 1
#else
#define HAS_ASYNC_LDS 0
#endif

__device__ __forceinline__ int imin(int a, int b) { return a < b ? a : b; }

// one 16-byte global -> LDS transfer (async DMA when available)
__device__ __forceinline__ void cp16_g2l(const float* __restrict__ g, float* l) {
#if HAS_ASYNC_LDS
    // integer round-trips avoid generic->named addrspace-cast restrictions;
    // low 32 bits of a generic LDS address are the LDS byte offset (ISA 10.2)
    __builtin_amdgcn_global_load_async_to_lds_b128(
        (g128p)(size_t)(const void*)g,
        (l128p)(unsigned)(size_t)(void*)l,
        0, 0);
#else
    *(f4*)l = *(const f4*)g;
#endif
}

template <int Ncnt>
__device__ __forceinline__ void wait_asynccnt_le() {
#if HAS_ASYNC_LDS
#if __has_builtin(__builtin_amdgcn_s_wait_asynccnt)
    __builtin_amdgcn_s_wait_asynccnt(Ncnt);
#else
    asm volatile("s_wait_asynccnt %0" :: "i"(Ncnt) : "memory");
#endif
#endif
}

__global__ __launch_bounds__(THREADS)
void knn_wmma_kernel(const float* __restrict__ samples,
                     const int*   __restrict__ targets,
                     const float* __restrict__ train,
                     const int*   __restrict__ labels,
                     float*       __restrict__ out)
{
    __shared__ float qsh[QB * QSTR];        // resident queries (reused as histogram later)
    __shared__ float gsh[2][NT * GSTR];     // double-buffered gallery chunk
    __shared__ float ssh[QB * SSTR];        // per-tile score tile
    __shared__ float tkV[QB][KNN];          // top-K values
    __shared__ int   tkI[QB][KNN];          // top-K gallery indices
    __shared__ float rsum[QB];

    const int tid   = threadIdx.x;
    const int lane  = tid & 31;
    const int wave  = tid >> 5;          // 0..7 -> N sub-tile
    const int half  = lane >> 4;         // 0 or 1 (K split per ISA layout)
    const int m16   = lane & 15;         // M (A) / N (B) index
    const int qbase = blockIdx.x * QB;

    // staging geometry: 16 float4 columns x 16 row-groups -> 8 issues/thread (= 8/wave)
    const int c4 = (tid & 15) << 2;      // float col within chunk (0..60)
    const int r0 = tid >> 4;             // 0..15

    // ---- load 16 resident query rows (f32) into LDS ----
    for (int i = tid * 4; i < QB * D; i += THREADS * 4) {
        const int r = i / D, col = i % D;
        *(f4*)&qsh[r * QSTR + col] = *(const f4*)&samples[(qbase + r) * D + col];
    }
    if (tid < QB) {
        for (int k = 0; k < KNN; ++k) { tkV[tid][k] = -__builtin_inff(); tkI[tid][k] = 0; }
    }
    float curMin = -__builtin_inff();
    int   minPos = 0;

    const int nTiles = (N + NT - 1) / NT;

    // issue one 128x64 gallery chunk (async DMA to LDS; 8 b128 per thread/wave)
    auto issueChunk = [&](int rbase, int kc, float* buf) {
        for (int rr = r0; rr < NT; rr += 16) {
            int gr = rbase + rr; if (gr >= N) gr = N - 1;
            cp16_g2l(&train[(size_t)gr * D + kc + c4], &buf[rr * GSTR + c4]);
            // warm L2 for the next tile's same chunk (global_prefetch_b8)
            int pr = gr + NT; if (pr >= N) pr = N - 1;
            __builtin_prefetch(&train[(size_t)pr * D + kc + c4], 0, 1);
        }
    };

    for (int tile = 0; tile < nTiles; ++tile) {
        const int rbase = tile * NT;

        v8f acc;
        #pragma unroll
        for (int r = 0; r < 8; ++r) acc[r] = 0.0f;

        // pipeline prologue: buf0 is free (all consumers passed last tile's barriers)
        issueChunk(rbase, 0, gsh[0]);

        for (int c = 0; c < NCH; ++c) {
            __syncthreads();   // (a) every wave done with buf[(c+1)&1] -> safe to refill
            if (c + 1 < NCH) {
                issueChunk(rbase, (c + 1) * DC, gsh[(c + 1) & 1]);
                wait_asynccnt_le<8>();   // chunk c landed (in-order), chunk c+1 in flight
            } else {
                wait_asynccnt_le<0>();
            }
            __syncthreads();   // (b) chunk c visible to all waves

            // ---- WMMA: 16 steps of 16x16x4 f32 over this chunk ----
            const float* qrow = &qsh[m16 * QSTR + c * DC + 2 * half];
            const float* grow = &gsh[c & 1][(wave * 16 + m16) * GSTR + 2 * half];
            #pragma unroll
            for (int kk = 0; kk < DC; kk += 4) {
                v2f a = *(const v2f*)(qrow + kk);   // A: M=m16, K=kk(+2*half)..+1
                v2f b = *(const v2f*)(grow + kk);   // B: N=row, K=kk(+2*half)..+1
                acc = __builtin_amdgcn_wmma_f32_16x16x4_f32(
                          false, a, false, b, (short)0, acc, false, false);
            }
        }

        __syncthreads();
        // ---- spill 16x16 score tile: C layout VGPR r -> M=r+8*half, N=m16 ----
        #pragma unroll
        for (int r = 0; r < 8; ++r)
            ssh[(r + 8 * half) * SSTR + wave * 16 + m16] = acc[r];
        __syncthreads();

        // ---- top-K update: one owner thread per query, replace-min ----
        if (tid < QB) {
            const int valid = imin(NT, N - rbase);
            for (int j = 0; j < valid; ++j) {
                const float v = ssh[tid * SSTR + j];
                if (v > curMin) {
                    tkV[tid][minPos] = v;
                    tkI[tid][minPos] = rbase + j;
                    float mn = tkV[tid][0]; int mp = 0;
                    for (int k = 1; k < KNN; ++k) {
                        const float x = tkV[tid][k];
                        if (x < mn) { mn = x; mp = k; }
                    }
                    curMin = mn; minPos = mp;
                }
            }
        }
    }

    // ---- class-vote histogram (reuse qsh as hist[QB][CLS]) ----
    __syncthreads();
    for (int i = tid; i < QB * CLS; i += THREADS) qsh[i] = 0.0f;
    __syncthreads();
    if (tid < QB) {
        float s = 0.0f;
        for (int k = 0; k < KNN; ++k) {
            const float v = tkV[tid][k];
            const int   c = labels[tkI[tid][k]];
            qsh[tid * CLS + c] += v;
            s += v;
        }
        rsum[tid] = (s == 0.0f) ? 1.0f : s;
    }
    __syncthreads();
    for (int i = tid; i < QB * CLS; i += THREADS) {
        const int q = i / CLS, c = i % CLS;
        out[(size_t)(qbase + q) * CLS + c] = qsh[i] / rsum[q];
    }
    // second tuple output: targets (as output dtype float)
    if (tid < QB)
        out[(size_t)Q * CLS + qbase + tid] = (float)targets[qbase + tid];
}

extern "C" void kernel_launch(void* const* d_in, const int* in_sizes, int n_in,
                              void* d_out, int out_size, void* d_ws, size_t ws_size,
                              hipStream_t stream) {
    (void)in_sizes; (void)n_in; (void)d_ws; (void)ws_size; (void)out_size;
    const float* samples = (const float*)d_in[0];
    const int*   targets = (const int*)  d_in[1];
    const float* train   = (const float*)d_in[2];
    const int*   labels  = (const int*)  d_in[3];
    float*       out     = (float*)d_out;

    knn_wmma_kernel<<<dim3(Q / QB), dim3(THREADS), 0, stream>>>(
        samples, targets, train, labels, out);
}